// DSBNSPDBNNet_50577534878051
// MI455X (gfx1250) — compile-verified
//
#include <hip/hip_runtime.h>

typedef float v2f __attribute__((ext_vector_type(2)));
typedef float v8f __attribute__((ext_vector_type(8)));

#define B_TOT   32768
#define N0      20
#define N1      15
#define ND      4
#define NCLS    4
#define ST0     33          // stride for 32-padded LDS matrices (gcd(33,64)=1)
#define ST1     17          // stride for 16-padded LDS matrices
#define EPS_REEIG 1e-4f
#define EPS_VAR   1e-5f
#define SQRT2_F   1.41421356237309515f

// ---------------- workspace layout (float offsets) ----------------
#define O_MEANL 0                         // ND*400 (atomic accum of logm sums)
#define O_CNT   (O_MEANL + ND*N0*N0)      // ND
#define O_GINV  (O_CNT + ND)              // ND*400
#define O_SIG2  (O_GINV + ND*N0*N0)       // ND (atomic)
#define O_SCALE (O_SIG2 + ND)             // ND
#define O_SLOGY (O_SCALE + ND)            // 225 (atomic)
#define O_GB    (O_SLOGY + N1*N1)         // 225
#define O_ACC_END 4096
#define O_LAM   O_ACC_END                 // B*20
#define O_U     (O_LAM + B_TOT*N0)        // B*300
#define O_Y     (O_U + B_TOT*N0*N1)       // B*225

// ---------------- wave-level LDS sync (cross-lane DS ordering) ----------------
__device__ __forceinline__ void lds_sync() {
  __builtin_amdgcn_wave_barrier();
  asm volatile("s_wait_dscnt 0" ::: "memory");
  __builtin_amdgcn_wave_barrier();
}

// ---------------- FP32 WMMA 16x16 tile matmul (fully compile-time specialized) ----
// D(16x16 at m0,nn0) = sum_k opA(A)[m][k] * (HASG?g[k]:1) * opB(B)[k][n]
// K multiple of 4; fragment mapping per CDNA5 ISA 7.12.2 (wave32).
template<int K, bool TA, bool TB, bool HASG>
__device__ __forceinline__ v8f wmma_tile(const float* A, int lda,
                                         const float* Bm, int ldb,
                                         int m0, int nn0,
                                         const float* g, int lane)
{
  v8f acc = {0.f,0.f,0.f,0.f,0.f,0.f,0.f,0.f};
  const int r  = lane & 15;
  const int hi = lane >> 4;
  const int ar = m0 + r;
  const int bc = nn0 + r;
#pragma unroll
  for (int k0 = 0; k0 < K; k0 += 4) {
    const int ka = k0 + 2*hi;
    v2f a, b;
    if (!TA) { a.x = A[ar*lda + ka];   a.y = A[ar*lda + ka + 1]; }
    else     { a.x = A[ka*lda + ar];   a.y = A[(ka+1)*lda + ar]; }
    if (HASG){ a.x *= g[ka];           a.y *= g[ka+1]; }
    if (!TB) { b.x = Bm[ka*ldb + bc];  b.y = Bm[(ka+1)*ldb + bc]; }
    else     { b.x = Bm[bc*ldb + ka];  b.y = Bm[bc*ldb + ka + 1]; }
    acc = __builtin_amdgcn_wmma_f32_16x16x4_f32(false, a, false, b,
                                                (short)0, acc, false, false);
  }
  return acc;
}

__device__ __forceinline__ void wmma_store_lds(float* C, int ldc, int m0, int nn0,
                                               v8f acc, int lane, int maxM, int maxN)
{
  const int col = nn0 + (lane & 15);
  const int rb  = m0 + (lane >> 4) * 8;
  if (col < maxN) {
#pragma unroll
    for (int v = 0; v < 8; ++v) {
      const int row = rb + v;
      if (row < maxM) C[row*ldc + col] = acc[v];
    }
  }
}

// ---------------- wave-parallel cyclic Jacobi eigensolver ----------------
// A: n x n symmetric in LDS (stride st, padded region zero). V must be zeroed by caller.
// On exit: wout[0..n-1] = eigenvalues (unsorted), V = eigenvectors (columns).
__device__ void jacobi(float* A, float* V, float* wout,
                       const int n, const int st, const int lane, const int sweeps)
{
  if (lane < n) V[lane*st + lane] = 1.f;
  lds_sync();
  for (int sw = 0; sw < sweeps; ++sw) {
    for (int p = 0; p < n - 1; ++p) {
      for (int q = p + 1; q < n; ++q) {
        const float apq = A[p*st + q];
        if (fabsf(apq) > 1e-12f) {
          const float app = A[p*st + p];
          const float aqq = A[q*st + q];
          const float tau = (aqq - app) / (2.0f * apq);
          const float t   = (tau >= 0.f ? 1.f : -1.f) /
                            (fabsf(tau) + sqrtf(1.f + tau*tau));
          const float c   = 1.0f / sqrtf(1.f + t*t);
          const float s   = t * c;
          if (lane < n) {                       // A <- A * J  (columns p,q)
            const float akp = A[lane*st + p], akq = A[lane*st + q];
            A[lane*st + p] = c*akp - s*akq;
            A[lane*st + q] = s*akp + c*akq;
          }
          lds_sync();
          if (lane < n) {                       // A <- J^T * A (rows p,q), V <- V*J
            const float apk = A[p*st + lane], aqk = A[q*st + lane];
            A[p*st + lane] = c*apk - s*aqk;
            A[q*st + lane] = s*apk + c*aqk;
            const float vkp = V[lane*st + p], vkq = V[lane*st + q];
            V[lane*st + p] = c*vkp - s*vkq;
            V[lane*st + q] = s*vkp + c*vkq;
          }
          lds_sync();
        }
      }
    }
  }
  if (lane < n) wout[lane] = A[lane*st + lane];
  lds_sync();
}

// ================= kernel 0: zero the accumulator region =================
__global__ void k_zero_acc(float* __restrict__ ws) {
  const int i = blockIdx.x * blockDim.x + threadIdx.x;
  if (i < O_ACC_END) ws[i] = 0.f;
}

// ================= kernel 1: L=logm(X), per-domain segment sums =================
#define WPB_A 4
__global__ __launch_bounds__(32*WPB_A)
void k_logm_accum(const float* __restrict__ X, const int* __restrict__ dom,
                  float* __restrict__ ws)
{
  __shared__ float sA[WPB_A][32*ST0];
  __shared__ float sV[WPB_A][32*ST0];
  __shared__ float sW[WPB_A][32];
  __shared__ float sG[WPB_A][32];
  __shared__ float dacc[ND][N0*N0];
  __shared__ float dcnt[ND];

  const int tid = threadIdx.x, lane = tid & 31, w = tid >> 5;
  for (int i = tid; i < ND*N0*N0; i += 32*WPB_A) dacc[i/(N0*N0)][i%(N0*N0)] = 0.f;
  if (tid < ND) dcnt[tid] = 0.f;
  __syncthreads();

  const int b = blockIdx.x * WPB_A + w;
  const int d = dom[b];
  float *A = sA[w], *V = sV[w];
  for (int i = lane; i < 32*ST0; i += 32) { A[i] = 0.f; V[i] = 0.f; }
  lds_sync();
  for (int i = lane; i < N0*N0; i += 32)
    A[(i/N0)*ST0 + (i%N0)] = X[(size_t)b*N0*N0 + i];
  lds_sync();

  jacobi(A, V, sW[w], N0, ST0, lane, 8);
  sG[w][lane] = (lane < N0) ? logf(sW[w][lane]) : 0.f;
  lds_sync();

  // L = V diag(log w) V^T, accumulate into block-local domain sums
  for (int mi = 0; mi < 32; mi += 16)
    for (int nj = 0; nj < 32; nj += 16) {
      v8f acc = wmma_tile<N0,false,true,true>(V, ST0, V, ST0, mi, nj, sG[w], lane);
      const int col = nj + (lane & 15);
      const int rb  = mi + (lane >> 4) * 8;
      if (col < N0) {
#pragma unroll
        for (int v = 0; v < 8; ++v) {
          const int row = rb + v;
          if (row < N0) atomicAdd(&dacc[d][row*N0 + col], acc[v]);
        }
      }
    }
  if (lane == 0) atomicAdd(&dcnt[d], 1.f);
  __syncthreads();
  for (int i = tid; i < ND*N0*N0; i += 32*WPB_A)
    atomicAdd(&ws[O_MEANL + i], dacc[i/(N0*N0)][i%(N0*N0)]);
  if (tid < ND) atomicAdd(&ws[O_CNT + tid], dcnt[tid]);
}

// ================= kernel 2: Ginv[d] = V e^{-w/2} V^T of meanL =================
__global__ __launch_bounds__(128)
void k_domain_whitener(float* __restrict__ ws)
{
  __shared__ float sA[ND][32*ST0];
  __shared__ float sV[ND][32*ST0];
  __shared__ float sW[ND][32];
  __shared__ float sG[ND][32];
  const int lane = threadIdx.x & 31, d = threadIdx.x >> 5;
  float *A = sA[d], *V = sV[d];
  for (int i = lane; i < 32*ST0; i += 32) { A[i] = 0.f; V[i] = 0.f; }
  lds_sync();
  const float inv = 1.0f / ws[O_CNT + d];
  for (int i = lane; i < N0*N0; i += 32)
    A[(i/N0)*ST0 + (i%N0)] = ws[O_MEANL + d*N0*N0 + i] * inv;
  lds_sync();
  jacobi(A, V, sW[d], N0, ST0, lane, 8);
  sG[d][lane] = (lane < N0) ? expf(-0.5f * sW[d][lane]) : 0.f;
  lds_sync();
  for (int mi = 0; mi < 32; mi += 16)
    for (int nj = 0; nj < 32; nj += 16) {
      v8f acc = wmma_tile<N0,false,true,true>(V, ST0, V, ST0, mi, nj, sG[d], lane);
      const int col = nj + (lane & 15);
      const int rb  = mi + (lane >> 4) * 8;
      if (col < N0) {
#pragma unroll
        for (int v = 0; v < 8; ++v) {
          const int row = rb + v;
          if (row < N0) ws[O_GINV + d*N0*N0 + row*N0 + col] = acc[v];
        }
      }
    }
}

// ======= kernel 3: M=sym(G X G), eigh(M), sig2 accum, store lambda & U=V^T W =======
#define WPB_C 4
__global__ __launch_bounds__(32*WPB_C)
void k_center_eig(const float* __restrict__ X, const int* __restrict__ dom,
                  const float* __restrict__ Wb, float* __restrict__ ws)
{
  __shared__ float s0[WPB_C][32*ST0];   // X, then M
  __shared__ float s1[WPB_C][32*ST0];   // G, then W
  __shared__ float s2[WPB_C][32*ST0];   // T = G@X, then V
  __shared__ float sW[WPB_C][32];
  const int tid = threadIdx.x, lane = tid & 31, w = tid >> 5;
  const int b = blockIdx.x * WPB_C + w;
  const int d = dom[b];
  float *Xm = s0[w], *Gm = s1[w], *Tm = s2[w];
  for (int i = lane; i < 32*ST0; i += 32) { Xm[i]=0.f; Gm[i]=0.f; Tm[i]=0.f; }
  lds_sync();
  for (int i = lane; i < N0*N0; i += 32) {
    Xm[(i/N0)*ST0 + (i%N0)] = X[(size_t)b*N0*N0 + i];
    Gm[(i/N0)*ST0 + (i%N0)] = ws[O_GINV + d*N0*N0 + i];
  }
  lds_sync();
  // T = G @ X
  for (int mi = 0; mi < 32; mi += 16)
    for (int nj = 0; nj < 32; nj += 16)
      wmma_store_lds(Tm, ST0, mi, nj,
                     wmma_tile<N0,false,false,false>(Gm, ST0, Xm, ST0, mi, nj, nullptr, lane),
                     lane, 32, 32);
  lds_sync();
  // M = T @ G -> Xm
  for (int mi = 0; mi < 32; mi += 16)
    for (int nj = 0; nj < 32; nj += 16)
      wmma_store_lds(Xm, ST0, mi, nj,
                     wmma_tile<N0,false,false,false>(Tm, ST0, Gm, ST0, mi, nj, nullptr, lane),
                     lane, 32, 32);
  lds_sync();
  // symmetrize
  for (int i = lane; i < N0*N0; i += 32) {
    const int r = i / N0, c = i % N0;
    if (r < c) {
      const float m = 0.5f * (Xm[r*ST0 + c] + Xm[c*ST0 + r]);
      Xm[r*ST0 + c] = m; Xm[c*ST0 + r] = m;
    }
  }
  lds_sync();
  for (int i = lane; i < 32*ST0; i += 32) Tm[i] = 0.f;   // V := 0
  lds_sync();
  jacobi(Xm, Tm, sW[w], N0, ST0, lane, 8);

  const float lam = (lane < N0) ? sW[w][lane] : 1.f;
  if (lane < N0) ws[O_LAM + (size_t)b*N0 + lane] = lam;
  const float gl = (lane < N0) ? logf(lam) : 0.f;
  float nr = gl * gl;                                    // ||Lc||_F^2 = sum log(w)^2
  for (int off = 16; off > 0; off >>= 1) nr += __shfl_xor(nr, off, 32);
  if (lane == 0) atomicAdd(&ws[O_SIG2 + d], nr);

  for (int i = lane; i < 32*ST0; i += 32) Gm[i] = 0.f;   // load W (20x15)
  lds_sync();
  for (int i = lane; i < N0*N1; i += 32)
    Gm[(i/N1)*ST0 + (i%N1)] = Wb[i];
  lds_sync();
  // U = V^T @ W  (20x15)
  for (int mi = 0; mi < 32; mi += 16) {
    v8f acc = wmma_tile<N0,true,false,false>(Tm, ST0, Gm, ST0, mi, 0, nullptr, lane);
    const int col = lane & 15;
    const int rb  = mi + (lane >> 4) * 8;
    if (col < N1) {
#pragma unroll
      for (int v = 0; v < 8; ++v) {
        const int row = rb + v;
        if (row < N0) ws[O_U + (size_t)b*(N0*N1) + row*N1 + col] = acc[v];
      }
    }
  }
}

// ================= kernel 4: scale[d] = s / sqrt(sig2/count + eps) =================
__global__ void k_scale(const float* __restrict__ sptr, float* __restrict__ ws)
{
  const int d = threadIdx.x;
  if (d < ND) {
    const float sig2 = ws[O_SIG2 + d] / ws[O_CNT + d];
    ws[O_SCALE + d] = sptr[0] / sqrtf(sig2 + EPS_VAR);
  }
}

// ===== kernel 5: Y = U^T diag(lam^scale) U; eigh(Y); accumulate sum logm(Y) =====
#define WPB_B 8
__global__ __launch_bounds__(32*WPB_B)
void k_bimap_logy(const int* __restrict__ dom, float* __restrict__ ws)
{
  __shared__ float sU[WPB_B][N0*16];
  __shared__ float sY[WPB_B][16*ST1];
  __shared__ float sQ[WPB_B][16*ST1];
  __shared__ float sMu[WPB_B][32];
  __shared__ float sG[WPB_B][32];
  __shared__ float accL[N1*N1];
  const int tid = threadIdx.x, lane = tid & 31, w = tid >> 5;
  for (int i = tid; i < N1*N1; i += 32*WPB_B) accL[i] = 0.f;
  __syncthreads();

  const int b = blockIdx.x * WPB_B + w;
  const int d = dom[b];
  float *U = sU[w], *Y = sY[w], *Q = sQ[w];
  for (int i = lane; i < N0*16; i += 32) U[i] = 0.f;
  for (int i = lane; i < 16*ST1; i += 32) { Y[i] = 0.f; Q[i] = 0.f; }
  sG[w][lane] = 0.f;
  lds_sync();
  for (int i = lane; i < N0*N1; i += 32)
    U[(i/N1)*16 + (i%N1)] = ws[O_U + (size_t)b*(N0*N1) + i];
  const float sc = ws[O_SCALE + d];
  if (lane < N0) {
    const float lam = ws[O_LAM + (size_t)b*N0 + lane];
    sG[w][lane] = expf(sc * logf(lam));                  // lam^scale
  }
  lds_sync();
  // Y[o][p] = sum_k U[k][o]*g[k]*U[k][p]
  v8f acc = wmma_tile<N0,true,false,true>(U, 16, U, 16, 0, 0, sG[w], lane);
  wmma_store_lds(Y, ST1, 0, 0, acc, lane, 16, 16);
  lds_sync();
  for (int i = lane; i < N1*N1; i += 32) {               // symmetrize
    const int r = i / N1, c = i % N1;
    if (r < c) {
      const float m = 0.5f * (Y[r*ST1 + c] + Y[c*ST1 + r]);
      Y[r*ST1 + c] = m; Y[c*ST1 + r] = m;
    }
  }
  lds_sync();
  for (int i = lane; i < N1*N1; i += 32)                 // store Y for kernel 7
    ws[O_Y + (size_t)b*(N1*N1) + i] = Y[(i/N1)*ST1 + (i%N1)];
  jacobi(Y, Q, sMu[w], N1, ST1, lane, 7);
  sG[w][lane] = (lane < N1) ? logf(sMu[w][lane]) : 0.f;
  lds_sync();
  // logm(Y) = Q diag(log mu) Q^T, accumulate
  v8f lacc = wmma_tile<16,false,true,true>(Q, ST1, Q, ST1, 0, 0, sG[w], lane);
  {
    const int col = lane & 15;
    const int rb  = (lane >> 4) * 8;
    if (col < N1) {
#pragma unroll
      for (int v = 0; v < 8; ++v) {
        const int row = rb + v;
        if (row < N1) atomicAdd(&accL[row*N1 + col], lacc[v]);
      }
    }
  }
  __syncthreads();
  for (int i = tid; i < N1*N1; i += 32*WPB_B)
    atomicAdd(&ws[O_SLOGY + i], accL[i]);
}

// ================= kernel 6: Gb = Q e^{-w/2} Q^T of mean logm(Y) =================
__global__ void k_gb(float* __restrict__ ws)
{
  __shared__ float A[16*ST1], Q[16*ST1], mu[32], g[32];
  const int lane = threadIdx.x;
  for (int i = lane; i < 16*ST1; i += 32) { A[i] = 0.f; Q[i] = 0.f; }
  lds_sync();
  const float invB = 1.0f / (float)B_TOT;
  for (int i = lane; i < N1*N1; i += 32)
    A[(i/N1)*ST1 + (i%N1)] = ws[O_SLOGY + i] * invB;
  lds_sync();
  jacobi(A, Q, mu, N1, ST1, lane, 8);
  g[lane] = (lane < N1) ? expf(-0.5f * mu[lane]) : 0.f;
  lds_sync();
  v8f acc = wmma_tile<16,false,true,true>(Q, ST1, Q, ST1, 0, 0, g, lane);
  const int col = lane & 15, rb = (lane >> 4) * 8;
  if (col < N1) {
#pragma unroll
    for (int v = 0; v < 8; ++v) {
      const int row = rb + v;
      if (row < N1) ws[O_GB + row*N1 + col] = acc[v];
    }
  }
}

// ===== kernel 7: Yn=sym(Gb Y Gb); ReEig+LogEig; half-vec; linear classifier =====
#define WPB_F 8
__global__ __launch_bounds__(32*WPB_F)
void k_final(const float* __restrict__ Wl, float* __restrict__ ws,
             float* __restrict__ out)
{
  __shared__ float sGb[16*ST1];
  __shared__ float sWl[NCLS*120];
  __shared__ float sY[WPB_F][16*ST1];
  __shared__ float sT[WPB_F][16*ST1];
  __shared__ float sQ[WPB_F][16*ST1];
  __shared__ float sMu[WPB_F][32];
  __shared__ float sG[WPB_F][32];
  const int tid = threadIdx.x, lane = tid & 31, w = tid >> 5;
  for (int i = tid; i < 16*ST1; i += 32*WPB_F) sGb[i] = 0.f;
  for (int i = tid; i < NCLS*120; i += 32*WPB_F) sWl[i] = Wl[i];
  __syncthreads();
  for (int i = tid; i < N1*N1; i += 32*WPB_F)
    sGb[(i/N1)*ST1 + (i%N1)] = ws[O_GB + i];
  __syncthreads();

  const int b = blockIdx.x * WPB_F + w;
  float *Y = sY[w], *T = sT[w], *Q = sQ[w];
  for (int i = lane; i < 16*ST1; i += 32) { Y[i]=0.f; T[i]=0.f; Q[i]=0.f; }
  lds_sync();
  for (int i = lane; i < N1*N1; i += 32)
    Y[(i/N1)*ST1 + (i%N1)] = ws[O_Y + (size_t)b*(N1*N1) + i];
  lds_sync();
  // T = Gb @ Y ; Yn = T @ Gb -> Y
  wmma_store_lds(T, ST1, 0, 0,
                 wmma_tile<16,false,false,false>(sGb, ST1, Y, ST1, 0, 0, nullptr, lane),
                 lane, 16, 16);
  lds_sync();
  wmma_store_lds(Y, ST1, 0, 0,
                 wmma_tile<16,false,false,false>(T, ST1, sGb, ST1, 0, 0, nullptr, lane),
                 lane, 16, 16);
  lds_sync();
  for (int i = lane; i < N1*N1; i += 32) {               // symmetrize
    const int r = i / N1, c = i % N1;
    if (r < c) {
      const float m = 0.5f * (Y[r*ST1 + c] + Y[c*ST1 + r]);
      Y[r*ST1 + c] = m; Y[c*ST1 + r] = m;
    }
  }
  lds_sync();
  jacobi(Y, Q, sMu[w], N1, ST1, lane, 7);
  // ReEig + LogEig fused: Lf = Q diag(log(max(mu,eps))) Q^T -> T
  sG[w][lane] = (lane < N1) ? logf(fmaxf(sMu[w][lane], EPS_REEIG)) : 0.f;
  lds_sync();
  wmma_store_lds(T, ST1, 0, 0,
                 wmma_tile<16,false,true,true>(Q, ST1, Q, ST1, 0, 0, sG[w], lane),
                 lane, 16, 16);
  lds_sync();
  // half-vectorize (15 diag, 105 * sqrt2 upper) and multiply by W_lin^T
  float pc0 = 0.f, pc1 = 0.f, pc2 = 0.f, pc3 = 0.f;
  for (int t = 0; t < 4; ++t) {
    const int e = lane + 32*t;
    if (e < 120) {
      float val;
      if (e < N1) {
        val = T[e*ST1 + e];
      } else {
        int t2 = e - N1, i = 0;
        while (t2 >= (N1 - 1 - i)) { t2 -= (N1 - 1 - i); ++i; }
        const int j = i + 1 + t2;
        val = SQRT2_F * T[i*ST1 + j];
      }
      pc0 += sWl[0*120 + e] * val;
      pc1 += sWl[1*120 + e] * val;
      pc2 += sWl[2*120 + e] * val;
      pc3 += sWl[3*120 + e] * val;
    }
  }
  for (int off = 16; off > 0; off >>= 1) {
    pc0 += __shfl_xor(pc0, off, 32);
    pc1 += __shfl_xor(pc1, off, 32);
    pc2 += __shfl_xor(pc2, off, 32);
    pc3 += __shfl_xor(pc3, off, 32);
  }
  if (lane == 0) {
    out[(size_t)b*NCLS + 0] = pc0;
    out[(size_t)b*NCLS + 1] = pc1;
    out[(size_t)b*NCLS + 2] = pc2;
    out[(size_t)b*NCLS + 3] = pc3;
  }
}

// ================================ launcher ================================
extern "C" void kernel_launch(void* const* d_in, const int* in_sizes, int n_in,
                              void* d_out, int out_size, void* d_ws, size_t ws_size,
                              hipStream_t stream)
{
  const float* X   = (const float*)d_in[0];   // [B,20,20]
  const int*   dom = (const int*)d_in[1];     // [B]
  const float* s   = (const float*)d_in[2];   // [1]
  const float* Wb  = (const float*)d_in[3];   // [20,15]
  const float* Wl  = (const float*)d_in[4];   // [4,120]
  float* ws  = (float*)d_ws;
  float* out = (float*)d_out;
  (void)in_sizes; (void)n_in; (void)out_size; (void)ws_size;

  hipLaunchKernelGGL(k_zero_acc,        dim3(O_ACC_END/256), dim3(256), 0, stream, ws);
  hipLaunchKernelGGL(k_logm_accum,      dim3(B_TOT/WPB_A), dim3(32*WPB_A), 0, stream, X, dom, ws);
  hipLaunchKernelGGL(k_domain_whitener, dim3(1), dim3(128), 0, stream, ws);
  hipLaunchKernelGGL(k_center_eig,      dim3(B_TOT/WPB_C), dim3(32*WPB_C), 0, stream, X, dom, Wb, ws);
  hipLaunchKernelGGL(k_scale,           dim3(1), dim3(32), 0, stream, s, ws);
  hipLaunchKernelGGL(k_bimap_logy,      dim3(B_TOT/WPB_B), dim3(32*WPB_B), 0, stream, dom, ws);
  hipLaunchKernelGGL(k_gb,              dim3(1), dim3(32), 0, stream, ws);
  hipLaunchKernelGGL(k_final,           dim3(B_TOT/WPB_F), dim3(32*WPB_F), 0, stream, Wl, ws, out);
}